// AdditiveAttention_69715909148735
// MI455X (gfx1250) — compile-verified
//
#include <hip/hip_runtime.h>
#include <math.h>

// Problem constants (from reference setup_inputs)
#define BB   4
#define TQ   256
#define TV   1024
#define DD   512
#define AAA  128
#define NEG_BIG (-1.0e9f)

typedef __attribute__((ext_vector_type(2))) float v2f;
typedef __attribute__((ext_vector_type(8))) float v8f;
typedef __attribute__((ext_vector_type(4))) unsigned int v4u;
typedef __attribute__((ext_vector_type(8))) int v8i;
typedef __attribute__((ext_vector_type(4))) int v4i;

// ---------------------------------------------------------------------------
// TDM: async-copy one 16x128 f32 tile (row stride 128 floats in global) into
// LDS with 4-DWORD padding after every 128 DWORDs (-> 132-float LDS pitch).
// D# layout per CDNA5 ISA ch. 8 (group0 128b, group1 256b, groups 2/3 zero
// for a 2D tensor). Tracked with TENSORcnt. 6-arg builtin form (clang-23).
// ---------------------------------------------------------------------------
__device__ __forceinline__ void tdm_load_k_tile(const float* gsrc, unsigned lds_addr)
{
    unsigned long long ga = (unsigned long long)(size_t)gsrc;
    v4u g0;
    g0[0] = 1u;                                            // count=1, user-mode D#
    g0[1] = lds_addr;                                      // lds_addr [63:32]
    g0[2] = (unsigned)(ga & 0xFFFFFFFFull);                // global_addr lo
    g0[3] = (unsigned)((ga >> 32) & 0x1FFFFFFull)          // global_addr hi (57-bit)
          | (2u << 30);                                    // type=2 ("image")
    v8i g1;
    g1[0] = (int)((2u << 16)      // data_size = 4 bytes
                | (1u << 20)      // pad_enable
                | (6u << 22)      // pad_interval: 128 DWORDs
                | (3u << 25));    // pad_amount:   4 DWORDs
    g1[1] = (int)(128u << 16);    // tensor_dim0 = 128 (lo16 @ bits 48..63)
    g1[2] = (int)(16u  << 16);    // tensor_dim0 hi = 0 | tensor_dim1 = 16 (lo16)
    g1[3] = (int)(128u << 16);    // tensor_dim1 hi = 0 | tile_dim0 = 128
    g1[4] = 16;                   // tile_dim1 = 16, tile_dim2 = 0
    g1[5] = 128;                  // tensor_dim0_stride = 128 (lo32)
    g1[6] = 0;                    // stride hi | tensor_dim1_stride lo (unused, 2D)
    g1[7] = 0;
    v4i z4 = {0, 0, 0, 0};
    v8i z8 = {0, 0, 0, 0, 0, 0, 0, 0};
    __builtin_amdgcn_tensor_load_to_lds(g0, g1, z4, z4, z8, 0);
}

// ---------------------------------------------------------------------------
// Kernel 1: fused projections  K = inputs@Wk + bk ; Q = context@Wq + bq
// One wave per 16x16 output tile, V_WMMA_F32_16X16X4_F32 over K-dim 512.
// ---------------------------------------------------------------------------
__global__ __launch_bounds__(256)
void proj_kernel(const float* __restrict__ inputs, const float* __restrict__ context,
                 const float* __restrict__ Wk, const float* __restrict__ bk,
                 const float* __restrict__ Wq, const float* __restrict__ bq,
                 float* __restrict__ Kws, float* __restrict__ Qws)
{
    int wave = blockIdx.x * 8 + (threadIdx.x >> 5);
    int lane = threadIdx.x & 31;
    int mt = wave >> 3;      // 0..319
    int nt = wave & 7;       // 0..7 (A=128 -> 8 tiles of 16)

    const float* X; const float* W; const float* bias; float* out;
    int rowbase;
    if (mt < 256) { X = inputs;  W = Wk; bias = bk; out = Kws; rowbase = mt * 16; }
    else          { X = context; W = Wq; bias = bq; out = Qws; rowbase = (mt - 256) * 16; }

    int halfsel = (lane >> 4) << 1;         // 0 for lanes 0-15, 2 for lanes 16-31
    int mrow = rowbase + (lane & 15);       // A-matrix: M = lane%16
    int ncol = nt * 16 + (lane & 15);       // B/C/D: N = lane%16

    v8f acc = {};
    #pragma unroll 4
    for (int k = 0; k < DD; k += 4) {
        v2f afrag, bfrag;
        const float* ap = X + (size_t)mrow * DD + k + halfsel;
        afrag[0] = ap[0];
        afrag[1] = ap[1];
        const float* bp = W + (size_t)(k + halfsel) * AAA + ncol;
        bfrag[0] = bp[0];
        bfrag[1] = bp[AAA];
        acc = __builtin_amdgcn_wmma_f32_16x16x4_f32(false, afrag, false, bfrag,
                                                    (short)0, acc, false, false);
    }

    float bbv = bias[ncol];
    int mhi = (lane >> 4) << 3;             // C/D rows: r / r+8 by lane half
    #pragma unroll
    for (int r = 0; r < 8; ++r) {
        out[(size_t)(rowbase + r + mhi) * AAA + ncol] = acc[r] + bbv;
    }
}

// ---------------------------------------------------------------------------
// Kernel 2: fused score (tanh dot) + online softmax + attn@V (WMMA f32)
// One block per (batch, 16-query tile). 8 waves; each wave owns a 64-wide
// D-slice of the accumulator. K chunks double-buffered via TDM async copy.
// ---------------------------------------------------------------------------
__global__ __launch_bounds__(256)
void attn_kernel(const float* __restrict__ inputs, const int* __restrict__ mask,
                 const float* __restrict__ attn_v,
                 const float* __restrict__ Kws, const float* __restrict__ Qws,
                 float* __restrict__ out)
{
    __shared__ float qbuf[16][132];      // 132-float pitch: conflict-free score loop
    __shared__ float kbuf[2][16][132];   // double-buffered TDM destination
    __shared__ float sbuf[16][16];
    __shared__ float pbuf[16][18];       // pad 18: conflict-free A-frag reads
    __shared__ float vv[AAA];
    __shared__ float mrow_s[16], lrow_s[16], crow_s[16];

    int b  = blockIdx.x >> 4;             // 16 q-tiles per batch
    int q0 = (blockIdx.x & 15) * 16;
    int tid  = threadIdx.x;
    int lane = tid & 31;
    int wav  = tid >> 5;

    const float* Kbase = Kws + (size_t)b * TV * AAA;

    // prologue: TDM-issue chunk 0 into buffer 0 (wave 0 only; EXEC-independent)
    if (wav == 0) {
        tdm_load_k_tile(Kbase, (unsigned)(size_t)&kbuf[0][0][0]);
    }

    // stage Q tile + attn_v, init softmax state (overlaps with TDM)
    for (int i = tid; i < 16 * AAA; i += 256) {
        int r = i >> 7, c = i & 127;
        qbuf[r][c] = Qws[(size_t)(b * TQ + q0 + r) * AAA + c];
    }
    if (tid < AAA) vv[tid] = attn_v[tid];
    if (tid < 16) { mrow_s[tid] = -3.0e38f; lrow_s[tid] = 0.0f; }

    int dbase = wav * 64;                 // this wave's D-slice
    v8f acc[4] = {};
    int qi = tid >> 4, vi = tid & 15;     // score-lattice assignment
    int halfsel = (lane >> 4) << 1;
    int n15 = lane & 15;
    int mhi = (lane >> 4) << 3;
    int cur = 0;

    for (int v0 = 0; v0 < TV; v0 += 16) {
        // ---- prefetch next K chunk (other buffer), wait for current ----
        if (wav == 0) {
            if (v0 + 16 < TV) {
                tdm_load_k_tile(Kbase + (size_t)(v0 + 16) * AAA,
                                (unsigned)(size_t)&kbuf[cur ^ 1][0][0]);
                __builtin_amdgcn_s_wait_tensorcnt(1);   // chunk v0 complete (in-order)
            } else {
                __builtin_amdgcn_s_wait_tensorcnt(0);
            }
        }
        __syncthreads();

        // ---- scores: s[q,v] = sum_a vv[a] * tanh(q[a] + k[a]) ----
        float s = 0.0f;
        #pragma unroll 4
        for (int a = 0; a < AAA; ++a) {
            s = fmaf(vv[a], tanhf(qbuf[qi][a] + kbuf[cur][vi][a]), s);
        }
        if (mask[b * TV + v0 + vi] == 0) s += NEG_BIG;
        sbuf[qi][vi] = s;
        __syncthreads();

        // ---- online softmax stats (16 threads, one per q-row) ----
        if (tid < 16) {
            int rq = tid;
            float mo = mrow_s[rq];
            float mn = mo;
            #pragma unroll
            for (int j = 0; j < 16; ++j) mn = fmaxf(mn, sbuf[rq][j]);
            float corr = __expf(mo - mn);
            float ls = 0.0f;
            #pragma unroll
            for (int j = 0; j < 16; ++j) {
                float p = __expf(sbuf[rq][j] - mn);
                pbuf[rq][j] = p;
                ls += p;
            }
            mrow_s[rq] = mn;
            lrow_s[rq] = lrow_s[rq] * corr + ls;
            crow_s[rq] = corr;
        }
        __syncthreads();

        // ---- rescale accumulators by softmax correction ----
        #pragma unroll
        for (int r = 0; r < 8; ++r) {
            float c = crow_s[r + mhi];
            acc[0][r] *= c; acc[1][r] *= c; acc[2][r] *= c; acc[3][r] *= c;
        }

        // ---- acc += p(16x16) @ V_chunk(16x64slice) via WMMA f32 16x16x4 ----
        const float* Vbase = inputs + (size_t)(b * TV + v0) * DD;
        #pragma unroll
        for (int kk = 0; kk < 16; kk += 4) {
            v2f afrag;
            afrag[0] = pbuf[n15][kk + halfsel];
            afrag[1] = pbuf[n15][kk + halfsel + 1];
            #pragma unroll
            for (int t = 0; t < 4; ++t) {
                int n = dbase + t * 16 + n15;
                v2f bfrag;
                bfrag[0] = Vbase[(size_t)(kk + halfsel) * DD + n];
                bfrag[1] = Vbase[(size_t)(kk + halfsel + 1) * DD + n];
                acc[t] = __builtin_amdgcn_wmma_f32_16x16x4_f32(false, afrag, false, bfrag,
                                                               (short)0, acc[t], false, false);
            }
        }
        __syncthreads();  // all reads of kbuf[cur^1-from-2-iters-ago]/pbuf done
        cur ^= 1;
    }

    // ---- normalize and store ----
    #pragma unroll
    for (int r = 0; r < 8; ++r) {
        float inv = 1.0f / lrow_s[r + mhi];
        #pragma unroll
        for (int t = 0; t < 4; ++t) {
            out[(size_t)(b * TQ + q0 + r + mhi) * DD + dbase + t * 16 + n15] = acc[t][r] * inv;
        }
    }
}

// ---------------------------------------------------------------------------
extern "C" void kernel_launch(void* const* d_in, const int* in_sizes, int n_in,
                              void* d_out, int out_size, void* d_ws, size_t ws_size,
                              hipStream_t stream)
{
    (void)in_sizes; (void)n_in; (void)out_size; (void)ws_size;
    const float* inputs  = (const float*)d_in[0];   // [B,Tv,D]
    const float* context = (const float*)d_in[1];   // [B,Tq,D]
    const int*   mask    = (const int*)  d_in[2];   // [B,Tv]
    const float* Wk      = (const float*)d_in[3];   // [D,A]
    const float* bk      = (const float*)d_in[4];   // [A]
    const float* Wq      = (const float*)d_in[5];   // [D,A]
    const float* bq      = (const float*)d_in[6];   // [A]
    const float* attn_v  = (const float*)d_in[7];   // [A]
    float* out = (float*)d_out;                     // [B,Tq,D]

    float* Kws = (float*)d_ws;                       // [B*Tv, A] = 2 MB
    float* Qws = Kws + (size_t)BB * TV * AAA;        // [B*Tq, A] = 0.5 MB

    // 2560 tiles / 8 waves-per-block = 320 blocks
    hipLaunchKernelGGL(proj_kernel, dim3(320), dim3(256), 0, stream,
                       inputs, context, Wk, bk, Wq, bq, Kws, Qws);
    // B * Tq/16 = 64 blocks
    hipLaunchKernelGGL(attn_kernel, dim3(BB * (TQ / 16)), dim3(256), 0, stream,
                       inputs, mask, attn_v, Kws, Qws, out);
}